// PointNetSetAbstraction_70239895159139
// MI455X (gfx1250) — compile-verified
//
#include <hip/hip_runtime.h>
#include <cstddef>

typedef __attribute__((ext_vector_type(2))) float v2f;
typedef __attribute__((ext_vector_type(8))) float v8f;

constexpr int BB  = 8;
constexpr int NN  = 8192;
constexpr int NPT = 2048;
constexpr int NS  = 32;
constexpr int DIN = 32;
constexpr int MROWS = BB * NPT * NS;   // 524288 GEMM rows
constexpr float R2 = 0.4f * 0.4f;

// ---------------------------------------------------------------------------
// Transpose xyz (B,3,N)->(B,N,3) and points (B,32,N)->(B,N,32) for gathers.
__global__ __launch_bounds__(256) void transpose_kernel(
    const float* __restrict__ xyz, const float* __restrict__ points,
    float* __restrict__ xyzT, float* __restrict__ ptT)
{
    int t = blockIdx.x * 256 + threadIdx.x;       // 0 .. B*N-1
    int b = t / NN, n = t - b * NN;
    #pragma unroll
    for (int c = 0; c < 3; c++)
        xyzT[(size_t)t * 3 + c] = xyz[((size_t)b * 3 + c) * NN + n];
    #pragma unroll
    for (int c = 0; c < DIN; c++)
        ptT[(size_t)t * DIN + c] = points[((size_t)b * DIN + c) * NN + n];
}

// ---------------------------------------------------------------------------
// Farthest point sampling: one block per batch, points resident in registers.
// Records `far` BEFORE the dist update (matches the jax scan semantics, so
// idx[0] == 0). Writes new_xyz_t (B,S,3) and the transposed output (B,3,S).
__global__ __launch_bounds__(1024) void fps_kernel(
    const float* __restrict__ xyz, int* __restrict__ fpsIdx,
    float* __restrict__ newxyz, float* __restrict__ out_newxyz)
{
    int b = blockIdx.x, tid = threadIdx.x;
    const float* xb = xyz + (size_t)b * 3 * NN;
    float px[8], py[8], pz[8], dist[8];
    #pragma unroll
    for (int i = 0; i < 8; i++) {
        int n = tid + i * 1024;
        px[i] = xb[n]; py[i] = xb[NN + n]; pz[i] = xb[2 * NN + n];
        dist[i] = 1e10f;
    }
    __shared__ float swv[32];
    __shared__ int   swi[32];
    __shared__ int   sfar;
    int far = 0;
    for (int s = 0; s < NPT; s++) {
        float cx = xb[far], cy = xb[NN + far], cz = xb[2 * NN + far];
        if (tid == 0) {
            fpsIdx[b * NPT + s] = far;
            newxyz[((size_t)b * NPT + s) * 3 + 0] = cx;
            newxyz[((size_t)b * NPT + s) * 3 + 1] = cy;
            newxyz[((size_t)b * NPT + s) * 3 + 2] = cz;
            out_newxyz[(size_t)b * 3 * NPT + 0 * NPT + s] = cx;
            out_newxyz[(size_t)b * 3 * NPT + 1 * NPT + s] = cy;
            out_newxyz[(size_t)b * 3 * NPT + 2 * NPT + s] = cz;
        }
        float bv = -1.0f; int bi = 0;
        #pragma unroll
        for (int i = 0; i < 8; i++) {
            float dx = px[i] - cx, dy = py[i] - cy, dz = pz[i] - cz;
            float d = dx * dx + dy * dy + dz * dz;
            d = fminf(dist[i], d);
            dist[i] = d;
            if (d > bv) { bv = d; bi = tid + i * 1024; }
        }
        // wave32 argmax (lowest index on ties)
        #pragma unroll
        for (int off = 16; off > 0; off >>= 1) {
            float ov = __shfl_down(bv, off);
            int   oi = __shfl_down(bi, off);
            if (ov > bv || (ov == bv && oi < bi)) { bv = ov; bi = oi; }
        }
        int wave = tid >> 5;
        if ((tid & 31) == 0) { swv[wave] = bv; swi[wave] = bi; }
        __syncthreads();
        if (wave == 0) {
            bv = swv[tid & 31]; bi = swi[tid & 31];
            #pragma unroll
            for (int off = 16; off > 0; off >>= 1) {
                float ov = __shfl_down(bv, off);
                int   oi = __shfl_down(bi, off);
                if (ov > bv || (ov == bv && oi < bi)) { bv = ov; bi = oi; }
            }
            if (tid == 0) sfar = bi;
        }
        __syncthreads();
        far = sfar;
    }
}

// ---------------------------------------------------------------------------
// Ball query: one wave per centroid; ballot + prefix popcount keeps the
// "first NS indices in ascending order" semantics with 32-way scanning.
__global__ __launch_bounds__(256) void ballq_kernel(
    const float* __restrict__ xyz, const float* __restrict__ newxyz,
    int* __restrict__ idx)
{
    int wave = threadIdx.x >> 5, lane = threadIdx.x & 31;
    int g = blockIdx.x * 8 + wave;          // centroid id = b*NPT + s
    int b = g >> 11;
    const float* xb = xyz + (size_t)b * 3 * NN;
    float cx = newxyz[(size_t)g * 3 + 0];
    float cy = newxyz[(size_t)g * 3 + 1];
    float cz = newxyz[(size_t)g * 3 + 2];
    int base = g * NS;
    int cnt = 0, first = NN - 1;            // empty -> clip(N, 0, N-1)
    for (int n0 = 0; n0 < NN; n0 += 32) {
        int n = n0 + lane;
        float dx = xb[n] - cx, dy = xb[NN + n] - cy, dz = xb[2 * NN + n] - cz;
        bool hit = (dx * dx + dy * dy + dz * dz) <= R2;
        unsigned long long mask = __ballot(hit);
        if (cnt == 0 && mask) first = n0 + __builtin_ctzll(mask);
        if (hit) {
            int pos = cnt + (int)__popcll(mask & ((1ull << lane) - 1ull));
            if (pos < NS) idx[base + pos] = n;
        }
        cnt += (int)__popcll(mask);
        if (cnt >= NS) break;               // wave-uniform
    }
    if (cnt < NS && lane >= cnt) idx[base + lane] = first;
}

// ---------------------------------------------------------------------------
// Layer-1 GEMM: rows = (b,s,k) grouped features (3 rel-xyz + 32 feat, pad->36),
// cols = 64. Block = 256 thr (8 waves, 2x4 tiles of 16x16), WMMA f32 16x16x4.
__global__ __launch_bounds__(256) void gemm1_kernel(
    const float* __restrict__ xyzT, const float* __restrict__ ptT,
    const float* __restrict__ newxyz, const int* __restrict__ idx,
    const float* __restrict__ w1, const float* __restrict__ b1,
    float* __restrict__ y1)
{
    __shared__ float xs[32][40];    // 32 rows x 36 (padded stride 40)
    __shared__ float bs[36][64];    // B matrix: bs[k][o] = w1[o][k]
    int tid = threadIdx.x;
    int row0 = blockIdx.x * 32;
    for (int e = tid; e < 36 * 64; e += 256) {
        int k = e >> 6, o = e & 63;
        bs[k][o] = (k < 35) ? w1[o * 35 + k] : 0.0f;
    }
    for (int e = tid; e < 32 * 36; e += 256) {
        int r = e / 36, c = e - r * 36;
        int R = row0 + r;
        int b = R >> 16;                    // / (NPT*NS)
        int rem = R & 65535;
        int s = rem >> 5, k = rem & 31;
        int j = idx[(b * NPT + s) * NS + k];
        float v;
        if (c < 3)       v = xyzT[((size_t)b * NN + j) * 3 + c]
                             - newxyz[((size_t)b * NPT + s) * 3 + c];
        else if (c < 35) v = ptT[((size_t)b * NN + j) * DIN + (c - 3)];
        else             v = 0.0f;
        xs[r][c] = v;
    }
    __syncthreads();
    int wave = tid >> 5, lane = tid & 31;
    int wm = wave >> 2, wn = wave & 3;
    int l16 = lane & 15;
    int kg = (lane >> 4) << 1;              // 0 or 2 (A/B K-split across half-waves)
    int m = wm * 16 + l16;
    int n = wn * 16 + l16;
    v8f acc = {};
    #pragma unroll
    for (int kk = 0; kk < 9; kk++) {
        int k0 = kk * 4 + kg;
        v2f a, bf;
        a.x = xs[m][k0];     a.y = xs[m][k0 + 1];
        bf.x = bs[k0][n];    bf.y = bs[k0 + 1][n];
        acc = __builtin_amdgcn_wmma_f32_16x16x4_f32(
            false, a, false, bf, (short)0, acc, false, false);
    }
    int col = wn * 16 + l16;
    float bias = b1[col];
    int rbase = row0 + wm * 16 + ((lane >> 4) << 3);
    #pragma unroll
    for (int v = 0; v < 8; v++)
        y1[(size_t)(rbase + v) * 64 + col] = acc[v] + bias;
}

// ---------------------------------------------------------------------------
// Layer-2 GEMM: applies folded BN1 + ReLU on the fly, K=64, N=128.
// Block = 512 thr (16 waves, 2x8 tiles of 16x16).
__global__ __launch_bounds__(512) void gemm2_kernel(
    const float* __restrict__ y1, const float* __restrict__ bn1,
    const float* __restrict__ w2, const float* __restrict__ b2,
    float* __restrict__ y2)
{
    __shared__ float xs[32][68];    // 32 rows x 64 (padded stride 68)
    __shared__ float bs[64][128];
    int tid = threadIdx.x;
    int row0 = blockIdx.x * 32;
    for (int e = tid; e < 64 * 128; e += 512) {
        int k = e >> 7, o = e & 127;
        bs[k][o] = w2[o * 64 + k];
    }
    const float* sc = bn1;
    const float* sh = bn1 + 64;
    for (int e = tid; e < 32 * 64; e += 512) {
        int r = e >> 6, c = e & 63;
        float v = y1[(size_t)(row0 + r) * 64 + c] * sc[c] + sh[c];
        xs[r][c] = fmaxf(v, 0.0f);
    }
    __syncthreads();
    int wave = tid >> 5, lane = tid & 31;
    int wm = wave >> 3, wn = wave & 7;
    int l16 = lane & 15;
    int kg = (lane >> 4) << 1;
    int m = wm * 16 + l16;
    int n = wn * 16 + l16;
    v8f acc = {};
    #pragma unroll
    for (int kk = 0; kk < 16; kk++) {
        int k0 = kk * 4 + kg;
        v2f a, bf;
        a.x = xs[m][k0];     a.y = xs[m][k0 + 1];
        bf.x = bs[k0][n];    bf.y = bs[k0 + 1][n];
        acc = __builtin_amdgcn_wmma_f32_16x16x4_f32(
            false, a, false, bf, (short)0, acc, false, false);
    }
    int col = wn * 16 + l16;
    float bias = b2[col];
    int rbase = row0 + wm * 16 + ((lane >> 4) << 3);
    #pragma unroll
    for (int v = 0; v < 8; v++)
        y2[(size_t)(rbase + v) * 128 + col] = acc[v] + bias;
}

// ---------------------------------------------------------------------------
// Per-channel sum / sum-of-squares partials (deterministic 2-stage reduce).
template <int C>
__global__ __launch_bounds__(256) void stats_kernel(
    const float* __restrict__ y, float* __restrict__ part, int rowsPerBlock)
{
    int tid = threadIdx.x;
    int c = tid & (C - 1);
    const int G = 256 / C;
    int rg = tid / C;
    size_t row0 = (size_t)blockIdx.x * rowsPerBlock;
    float s = 0.0f, q = 0.0f;
    for (int r = rg; r < rowsPerBlock; r += G) {
        float v = y[(row0 + r) * C + c];
        s += v; q += v * v;
    }
    __shared__ float ss[256], sq[256];
    ss[tid] = s; sq[tid] = q;
    __syncthreads();
    if (tid < C) {
        for (int g = 1; g < G; g++) { s += ss[tid + g * C]; q += sq[tid + g * C]; }
        part[blockIdx.x * C + tid] = s;
        part[256 * C + blockIdx.x * C + tid] = q;
    }
}

// Fold BN (biased var, eps=1e-5) into scale/shift: y*scale + shift.
template <int C>
__global__ void finalize_kernel(
    const float* __restrict__ part, const float* __restrict__ g,
    const float* __restrict__ be, float* __restrict__ bn)
{
    int c = threadIdx.x;
    float s = 0.0f, q = 0.0f;
    for (int b = 0; b < 256; b++) { s += part[b * C + c]; q += part[256 * C + b * C + c]; }
    const float M = (float)MROWS;
    float mean = s / M;
    float var  = q / M - mean * mean;
    float scale = g[c] * rsqrtf(var + 1e-5f);
    bn[c]     = scale;
    bn[C + c] = be[c] - mean * scale;
}

// ---------------------------------------------------------------------------
// BN2 + ReLU + max over the NS samples; write transposed (B,128,S) output.
__global__ __launch_bounds__(128) void maxpool_kernel(
    const float* __restrict__ y2, const float* __restrict__ bn2,
    float* __restrict__ out)
{
    int c = threadIdx.x;
    int bs = blockIdx.x;                    // b*NPT + s
    float sc = bn2[c], sh = bn2[128 + c];
    size_t base = (size_t)bs * NS * 128;
    float m = 0.0f;                         // relu output >= 0
    #pragma unroll
    for (int k = 0; k < NS; k++) {
        float v = y2[base + (size_t)k * 128 + c] * sc + sh;
        m = fmaxf(m, fmaxf(v, 0.0f));
    }
    int b = bs >> 11, s = bs & (NPT - 1);
    out[(size_t)b * 128 * NPT + (size_t)c * NPT + s] = m;
}

// ---------------------------------------------------------------------------
extern "C" void kernel_launch(void* const* d_in, const int* in_sizes, int n_in,
                              void* d_out, int out_size, void* d_ws, size_t ws_size,
                              hipStream_t stream)
{
    (void)in_sizes; (void)n_in; (void)out_size; (void)ws_size;
    const float* xyz    = (const float*)d_in[0];
    const float* points = (const float*)d_in[1];
    const float* w1  = (const float*)d_in[2];
    const float* b1  = (const float*)d_in[3];
    const float* g1  = (const float*)d_in[4];
    const float* be1 = (const float*)d_in[5];
    const float* w2  = (const float*)d_in[6];
    const float* b2  = (const float*)d_in[7];
    const float* g2  = (const float*)d_in[8];
    const float* be2 = (const float*)d_in[9];
    float* out = (float*)d_out;

    char* p = (char*)d_ws;
    auto carve = [&](size_t bytes) -> char* {
        char* r = p;
        p += (bytes + 255) & ~(size_t)255;
        return r;
    };
    float* xyzT   = (float*)carve(sizeof(float) * (size_t)BB * NN * 3);
    float* ptT    = (float*)carve(sizeof(float) * (size_t)BB * NN * DIN);
    int*   fpsIdx = (int*)  carve(sizeof(int)   * (size_t)BB * NPT);
    float* newxyz = (float*)carve(sizeof(float) * (size_t)BB * NPT * 3);
    int*   idx    = (int*)  carve(sizeof(int)   * (size_t)BB * NPT * NS);
    float* y1     = (float*)carve(sizeof(float) * (size_t)MROWS * 64);
    float* y2     = (float*)carve(sizeof(float) * (size_t)MROWS * 128);
    float* part1  = (float*)carve(sizeof(float) * 2 * 256 * 64);
    float* part2  = (float*)carve(sizeof(float) * 2 * 256 * 128);
    float* bn1    = (float*)carve(sizeof(float) * 128);
    float* bn2    = (float*)carve(sizeof(float) * 256);

    transpose_kernel<<<BB * NN / 256, 256, 0, stream>>>(xyz, points, xyzT, ptT);
    fps_kernel<<<BB, 1024, 0, stream>>>(xyz, fpsIdx, newxyz, out);
    ballq_kernel<<<BB * NPT / 8, 256, 0, stream>>>(xyz, newxyz, idx);
    gemm1_kernel<<<MROWS / 32, 256, 0, stream>>>(xyzT, ptT, newxyz, idx, w1, b1, y1);
    stats_kernel<64><<<256, 256, 0, stream>>>(y1, part1, MROWS / 256);
    finalize_kernel<64><<<1, 64, 0, stream>>>(part1, g1, be1, bn1);
    gemm2_kernel<<<MROWS / 32, 512, 0, stream>>>(y1, bn1, w2, b2, y2);
    stats_kernel<128><<<256, 256, 0, stream>>>(y2, part2, MROWS / 256);
    finalize_kernel<128><<<1, 128, 0, stream>>>(part2, g2, be2, bn2);
    maxpool_kernel<<<BB * NPT, 128, 0, stream>>>(y2, bn2, out + (size_t)BB * 3 * NPT);
}